// Copy_Decoder_48876727829077
// MI455X (gfx1250) — compile-verified
//
#include <hip/hip_runtime.h>
#include <math.h>

#define H   1024
#define L   64
#define S   64
#define V   32000
#define EMB (H + L)        // 1088
#define GIN (2*H + L)      // 2112
#define G3H (3*H)          // 3072

typedef float v2f __attribute__((ext_vector_type(2)));
typedef float v8f __attribute__((ext_vector_type(8)));

// workspace layout (float offsets)
#define WS_X     0          // x[2112]
#define WS_JOIN  2112       // joined[2048]: [0:1024]=h_new, [1024:2048]=attn_ctx
#define WS_GI    4160       // gi[3072]
#define WS_GH    7232       // gh[3072]
#define WS_SG    10304      // score_g[32000]
#define WS_SC    42304      // score_c[64]
#define WS_RED   42368      // {max, log-sum-exp}

// output layout: [0:32000) probs | [32000:33024) attn_context |
//                [33024:34048) current_hidden | [34048:34112) attn_weights

// ---------------- k1: build x = [loc, emb_tail, prev_context]; zero score_c --
__global__ void k1_build_x(const int* word_input, const float* prev_context,
                           const int* sources, const int* targets,
                           const int* ti_p, const int* utf_p,
                           const float* emb_table, float* ws) {
  int i = blockIdx.x * blockDim.x + threadIdx.x;
  if (blockIdx.x == 0 && threadIdx.x < S) ws[WS_SC + threadIdx.x] = 0.0f;
  if (i >= GIN) return;
  float* x = ws + WS_X;
  int w = word_input[0];
  if (i < S) {
    int di = utf_p[0] ? targets[ti_p[0]] : w;
    int cnt = 0;
    for (int s = 0; s < S; ++s) cnt += (sources[s] == di) ? 1 : 0;
    float match = (sources[i] == di) ? 1.0f : 0.0f;
    float nrm = fmaxf(sqrtf((float)cnt), 1e-12f);
    x[i] = match / nrm;
  } else if (i < EMB) {
    x[i] = emb_table[(long long)w * EMB + i];
  } else {
    x[i] = prev_context[i - EMB];
  }
}

// ---------------- k2: GRU GEMVs gi = Wih@x + bih, gh = Whh@h + bhh -----------
// wave-per-row, float4 (b128) streaming loads, wave32 shuffle reduction
__global__ void k2_gru_gemv(const float* __restrict__ Wih,
                            const float* __restrict__ Whh,
                            const float* __restrict__ bih,
                            const float* __restrict__ bhh,
                            const float* __restrict__ prev_hidden,
                            float* __restrict__ ws) {
  const int lane = threadIdx.x & 31;
  const int wid  = blockIdx.x * (blockDim.x >> 5) + (threadIdx.x >> 5);
  float acc = 0.f;
  if (wid < G3H) {
    const float4* row = (const float4*)(Wih + (long long)wid * GIN);
    const float4* xv  = (const float4*)(ws + WS_X);
    for (int k = lane; k < GIN/4; k += 32) {
      float4 a = row[k], b = xv[k];
      acc += a.x*b.x + a.y*b.y + a.z*b.z + a.w*b.w;
    }
  } else {
    const int r = wid - G3H;
    const float4* row = (const float4*)(Whh + (long long)r * H);
    const float4* hv  = (const float4*)prev_hidden;
    for (int k = lane; k < H/4; k += 32) {
      float4 a = row[k], b = hv[k];
      acc += a.x*b.x + a.y*b.y + a.z*b.z + a.w*b.w;
    }
  }
  for (int off = 16; off; off >>= 1) acc += __shfl_xor(acc, off, 32);
  if (lane == 0) {
    if (wid < G3H) ws[WS_GI + wid] = acc + bih[wid];
    else           ws[WS_GH + (wid - G3H)] = acc + bhh[wid - G3H];
  }
}

// ---------------- k3: GRU gates -> h_new ------------------------------------
__global__ void k3_gates(const float* prev_hidden, float* ws, float* out) {
  int i = blockIdx.x * blockDim.x + threadIdx.x;
  if (i >= H) return;
  const float* gi = ws + WS_GI;
  const float* gh = ws + WS_GH;
  float r = 1.f / (1.f + expf(-(gi[i] + gh[i])));
  float z = 1.f / (1.f + expf(-(gi[H+i] + gh[H+i])));
  float n = tanhf(gi[2*H+i] + r * gh[2*H+i]);
  float hn = (1.f - z) * n + z * prev_hidden[i];
  ws[WS_JOIN + i] = hn;        // joined[0:H]
  out[V + H + i]  = hn;        // current_hidden
}

// ---------------- k4: attention (single block) ------------------------------
__global__ void k4_attn(const float* __restrict__ enc, float* ws, float* out) {
  __shared__ float lg[S];
  __shared__ float st[2];
  const int tid = threadIdx.x, lane = tid & 31, wv = tid >> 5;  // 8 waves
  const float* hn = ws + WS_JOIN;
  for (int s = wv*8; s < wv*8 + 8; ++s) {
    float acc = 0.f;
    const float* er = enc + (long long)s * H;
    for (int k = lane; k < H; k += 32) acc += er[k] * hn[k];
    for (int off = 16; off; off >>= 1) acc += __shfl_xor(acc, off, 32);
    if (lane == 0) lg[s] = acc;
  }
  __syncthreads();
  if (tid == 0) {
    float m = lg[0];
    for (int s = 1; s < S; ++s) m = fmaxf(m, lg[s]);
    float d = 0.f;
    for (int s = 0; s < S; ++s) d += expf(lg[s] - m);
    st[0] = m; st[1] = d;
  }
  __syncthreads();
  if (tid < S) {
    float wgt = expf(lg[tid] - st[0]) / st[1];
    lg[tid] = wgt;
    out[V + 2*H + tid] = wgt;    // attn_weights
  }
  __syncthreads();
  for (int j = tid; j < H; j += blockDim.x) {
    float acc = 0.f;
    for (int s = 0; s < S; ++s) acc += lg[s] * enc[(long long)s * H + j];
    ws[WS_JOIN + H + j] = acc;   // joined[H:2H]
    out[V + j] = acc;            // attn_context
  }
}

// ---------------- k5: score_g = gen_W @ joined + gen_b (262 MB stream) ------
__global__ void k5_gen_gemv(const float* __restrict__ gen_W,
                            const float* __restrict__ gen_b,
                            float* __restrict__ ws) {
  const int lane = threadIdx.x & 31;
  const int wid  = blockIdx.x * (blockDim.x >> 5) + (threadIdx.x >> 5);
  if (wid >= V) return;
  const float4* row = (const float4*)(gen_W + (long long)wid * (2*H));
  const float4* jv  = (const float4*)(ws + WS_JOIN);
  float acc = 0.f;
  for (int k = lane; k < (2*H)/4; k += 32) {
    __builtin_prefetch(&row[k + 64], 0, 0);   // global_prefetch_b8, ~1KB ahead
    float4 a = row[k]; float4 b = jv[k];
    acc += a.x*b.x + a.y*b.y + a.z*b.z + a.w*b.w;
  }
  for (int off = 16; off; off >>= 1) acc += __shfl_xor(acc, off, 32);
  if (lane == 0) ws[WS_SG + wid] = acc + gen_b[wid];
}

// ---------------- k6: WMMA fp32 GEMM: score_c via tanh(enc@W^T+b) @ hprev ---
// D(16x16) tile per wave; A = enc tile (16x4), B = copy_W^T tile (4x16).
// Lane layout per CDNA5 ISA: lanes 0-15 hold K={k,k+1}, lanes 16-31 K={k+2,k+3}.
__global__ void __launch_bounds__(128)
k6_copy_score(const float* __restrict__ enc, const float* __restrict__ copy_W,
              const float* __restrict__ copy_b,
              const float* __restrict__ prev_hidden, float* __restrict__ ws) {
  const int lane  = threadIdx.x & 31;
  const int wv    = threadIdx.x >> 5;      // 0..3 -> M tile
  const int jBase = blockIdx.x * 16;       // N tile
  const int sBase = wv * 16;
  const int half  = lane >> 4;
  const int l16   = lane & 15;
  v8f acc = {};
  const float* aRow = enc    + (long long)(sBase + l16) * H;  // A row m=l16
  const float* bRow = copy_W + (long long)(jBase + l16) * H;  // B col n=l16
  for (int k = 0; k < H; k += 4) {
    const int kk = k + 2*half;
    v2f a = *(const v2f*)(aRow + kk);
    v2f b = *(const v2f*)(bRow + kk);
    acc = __builtin_amdgcn_wmma_f32_16x16x4_f32(false, a, false, b,
                                                (short)0, acc, false, false);
  }
  // epilogue: tanh(acc + b[j]) * hprev[j], reduce over j (16 lanes/half-wave)
  const int j = jBase + l16;
  const float cb = copy_b[j];
  const float hp = prev_hidden[j];
  for (int r = 0; r < 8; ++r) {
    float val = tanhf(acc[r] + cb) * hp;
    for (int off = 1; off < 16; off <<= 1) val += __shfl_xor(val, off, 32);
    if (l16 == 0) {
      int m = r + 8*half;                       // row within tile
      atomicAdd(&ws[WS_SC + sBase + m], val);
    }
  }
}

// ---------------- k7a: log-sum-exp over [score_g, score_c] ------------------
__global__ void k7a_lse(float* ws) {
  __shared__ float sd[256];
  const int tid = threadIdx.x;
  const float* sg = ws + WS_SG;
  const float* sc = ws + WS_SC;
  float m = -3.0e38f;
  for (int v = tid; v < V + S; v += 256) {
    float x = (v < V) ? sg[v] : sc[v - V];
    m = fmaxf(m, x);
  }
  sd[tid] = m; __syncthreads();
  for (int stp = 128; stp; stp >>= 1) {
    if (tid < stp) sd[tid] = fmaxf(sd[tid], sd[tid + stp]);
    __syncthreads();
  }
  m = sd[0]; __syncthreads();
  float sum = 0.f;
  for (int v = tid; v < V + S; v += 256) {
    float x = (v < V) ? sg[v] : sc[v - V];
    sum += expf(x - m);
  }
  sd[tid] = sum; __syncthreads();
  for (int stp = 128; stp; stp >>= 1) {
    if (tid < stp) sd[tid] += sd[tid + stp];
    __syncthreads();
  }
  if (tid == 0) { ws[WS_RED] = m; ws[WS_RED + 1] = logf(sd[0]); }
}

// ---------------- k7b: output = prob_g --------------------------------------
__global__ void k7b_write(const float* ws, float* out) {
  int v = blockIdx.x * blockDim.x + threadIdx.x;
  if (v >= V) return;
  out[v] = ws[WS_SG + v] - ws[WS_RED] - ws[WS_RED + 1];
}

// ---------------- k7c: scatter-add prob_c at sources ------------------------
__global__ void k7c_scatter(const int* sources, const float* ws, float* out) {
  int s = threadIdx.x;
  if (s < S) {
    float pc = ws[WS_SC + s] - ws[WS_RED] - ws[WS_RED + 1];
    atomicAdd(&out[sources[s]], pc);
  }
}

extern "C" void kernel_launch(void* const* d_in, const int* in_sizes, int n_in,
                              void* d_out, int out_size, void* d_ws, size_t ws_size,
                              hipStream_t stream) {
  const int*   word_input  = (const int*)  d_in[0];
  const float* prev_ctx    = (const float*)d_in[1];
  const float* prev_hidden = (const float*)d_in[2];
  const float* enc         = (const float*)d_in[3];
  const int*   sources     = (const int*)  d_in[4];
  const int*   targets     = (const int*)  d_in[5];
  const int*   ti          = (const int*)  d_in[6];
  const int*   utf         = (const int*)  d_in[7];
  const float* emb         = (const float*)d_in[8];
  const float* Wih         = (const float*)d_in[9];
  const float* Whh         = (const float*)d_in[10];
  const float* bih         = (const float*)d_in[11];
  const float* bhh         = (const float*)d_in[12];
  const float* copy_W      = (const float*)d_in[13];
  const float* copy_b      = (const float*)d_in[14];
  const float* gen_W       = (const float*)d_in[15];
  const float* gen_b       = (const float*)d_in[16];
  float* out = (float*)d_out;
  float* ws  = (float*)d_ws;

  k1_build_x <<<(GIN + 255)/256, 256, 0, stream>>>(word_input, prev_ctx, sources,
                                                   targets, ti, utf, emb, ws);
  k2_gru_gemv<<<(2*G3H)/8, 256, 0, stream>>>(Wih, Whh, bih, bhh, prev_hidden, ws);
  k3_gates   <<<H/256, 256, 0, stream>>>(prev_hidden, ws, out);
  k4_attn    <<<1, 256, 0, stream>>>(enc, ws, out);
  k5_gen_gemv<<<V/8, 256, 0, stream>>>(gen_W, gen_b, ws);
  k6_copy_score<<<H/16, 128, 0, stream>>>(enc, copy_W, copy_b, prev_hidden, ws);
  k7a_lse    <<<1, 256, 0, stream>>>(ws);
  k7b_write  <<<V/256, 256, 0, stream>>>(ws, out);
  k7c_scatter<<<1, 64, 0, stream>>>(sources, ws, out);
}